// ChaoticLSTM_31662498906897
// MI455X (gfx1250) — compile-verified
//
#include <hip/hip_runtime.h>

// Problem constants (from reference setup_inputs)
#define BS   64
#define SEQ  512
#define IN   256
#define HID  512
#define GH   2048   // 4*HID

typedef __attribute__((ext_vector_type(2))) float v2f;
typedef __attribute__((ext_vector_type(8))) float v8f;

// ---- fast scalar math on hardware trans units -------------------------------
__device__ __forceinline__ float fast_rcp(float x) { return __builtin_amdgcn_rcpf(x); }
__device__ __forceinline__ float fast_sig(float x) { return fast_rcp(1.0f + __expf(-x)); }
__device__ __forceinline__ float fast_tanh(float x) {
    // tanh(x) = 2*sigmoid(2x) - 1
    return fmaf(2.0f, fast_rcp(1.0f + __expf(-2.0f * x)), -1.0f);
}

// ---- chaotic Lee oscillator activation --------------------------------------
// z_{n+1} = (f(0.6u+0.6v-0.5z+0.5x) - f(-0.6u-0.6v-0.5z+0.5x)) * e^{-50x^2} + f(x)
template <bool TANH>
__device__ __forceinline__ float lee(float x) {
    float w     = TANH ? fast_tanh(x) : fast_sig(x);
    float decay = __expf(-50.0f * x * x);
    float u = 0.0f, v = 0.0f, z = 0.0f;
    #pragma unroll 5
    for (int i = 0; i < 25; ++i) {
        float a  = 0.6f * (u + v);
        float ct = fmaf(-0.5f, z, 0.5f * x);
        float u1 = TANH ? fast_tanh(a + ct) : fast_sig(a + ct);
        float v1 = TANH ? fast_tanh(ct - a) : fast_sig(ct - a);
        z = fmaf(u1 - v1, decay, w);
        u = u1; v = v1;
    }
    return z;
}

// ---- per-step gate GEMM + Lee activation ------------------------------------
// One wave computes one 16x16 tile of gates[64][2048] with f32 WMMA (16x16x4),
// K = 256 (xt@Wi) + 512 (h@Wh), then applies the Lee oscillator in-register.
// A layout (16x4 f32): lane l holds A[l&15][2*(l>>4) + {0,1}]
// B layout (4x16 f32): lane l holds B[2*(l>>4) + {0,1}][l&15]
// C layout: VGPR r -> row (r + 8*(l>>4)), col (l&15); same col for all 8 rows.
__global__ __launch_bounds__(128) void gates_kernel(
    const float* __restrict__ x, const float* __restrict__ Wi,
    const float* __restrict__ Wh, const float* __restrict__ bias,
    const float* __restrict__ hbuf, float* __restrict__ act, int t)
{
    const int wave = blockIdx.x * (blockDim.x >> 5) + (threadIdx.x >> 5); // 0..511
    const int lane = threadIdx.x & 31;
    const int mt   = wave & 3;    // batch tile  (4 x 16 rows)
    const int nt   = wave >> 2;   // column tile (128 x 16 cols)
    const int half = lane >> 4;   // 0 or 1
    const int l15  = lane & 15;
    const int arow = mt * 16 + l15;   // A-matrix row (batch index)
    const int ncol = nt * 16 + l15;   // B/C column (gate index)

    v8f acc = {};

    // ---- xt @ Wi, K = 256 ----
    const float* xrow = x + ((size_t)arow * SEQ + t) * IN;
    for (int k = 0; k < IN; k += 4) {
        const int ka = k + 2 * half;
        v2f a;  a.x  = xrow[ka];
                a.y  = xrow[ka + 1];
        v2f bb; bb.x = Wi[(size_t)ka * GH + ncol];
                bb.y = Wi[(size_t)(ka + 1) * GH + ncol];
        acc = __builtin_amdgcn_wmma_f32_16x16x4_f32(false, a, false, bb,
                                                    (short)0, acc, false, false);
    }

    // warm the Wh stream (global_prefetch_b8)
    __builtin_prefetch(&Wh[(size_t)(2 * half) * GH + ncol], 0, 1);

    // ---- h @ Wh, K = 512 ----
    const float* hrow = hbuf + (size_t)arow * HID;
    for (int k = 0; k < HID; k += 4) {
        const int ka = k + 2 * half;
        v2f a;  a.x  = hrow[ka];
                a.y  = hrow[ka + 1];
        v2f bb; bb.x = Wh[(size_t)ka * GH + ncol];
                bb.y = Wh[(size_t)(ka + 1) * GH + ncol];
        acc = __builtin_amdgcn_wmma_f32_16x16x4_f32(false, a, false, bb,
                                                    (short)0, acc, false, false);
    }

    // ---- bias + Lee oscillator, store activated gates ----
    const float bn   = bias[ncol];
    const int   gate = nt >> 5;             // 0:i 1:f 2:g 3:o  (wave-uniform)
    const int   row0 = mt * 16 + half * 8;  // C-layout row base for this lane
    if (gate == 2) {
        #pragma unroll
        for (int r = 0; r < 8; ++r)
            act[(size_t)(row0 + r) * GH + ncol] = lee<true>(acc[r] + bn);
    } else {
        #pragma unroll
        for (int r = 0; r < 8; ++r)
            act[(size_t)(row0 + r) * GH + ncol] = lee<false>(acc[r] + bn);
    }
}

// ---- per-step cell/hidden update --------------------------------------------
__global__ __launch_bounds__(256) void cell_kernel(
    const float* __restrict__ act, float* __restrict__ cbuf,
    float* __restrict__ hbuf, float* __restrict__ out, int t)
{
    const int idx = blockIdx.x * blockDim.x + threadIdx.x;  // 0..BS*HID-1
    const int b   = idx >> 9;        // / HID
    const int hc  = idx & (HID - 1);
    const float* ga = act + (size_t)b * GH;
    const float it = ga[hc];
    const float ft = ga[HID + hc];
    const float gt = ga[2 * HID + hc];
    const float ot = ga[3 * HID + hc];
    const float c1 = fmaf(ft, cbuf[idx], it * gt);
    cbuf[idx] = c1;
    const float h1 = ot * lee<true>(c1);
    hbuf[idx] = h1;
    out[((size_t)b * SEQ + t) * HID + hc] = h1;
}

extern "C" void kernel_launch(void* const* d_in, const int* in_sizes, int n_in,
                              void* d_out, int out_size, void* d_ws, size_t ws_size,
                              hipStream_t stream) {
    const float* x    = (const float*)d_in[0];
    const float* Wi   = (const float*)d_in[1];
    const float* Wh   = (const float*)d_in[2];
    const float* bias = (const float*)d_in[3];
    const float* h0   = (const float*)d_in[4];
    const float* c0   = (const float*)d_in[5];
    float* out = (float*)d_out;

    // workspace: activated gates [64][2048], h [64][512], c [64][512]
    float* act  = (float*)d_ws;
    float* hbuf = act + (size_t)BS * GH;
    float* cbuf = hbuf + (size_t)BS * HID;

    hipMemcpyAsync(hbuf, h0, (size_t)BS * HID * sizeof(float),
                   hipMemcpyDeviceToDevice, stream);
    hipMemcpyAsync(cbuf, c0, (size_t)BS * HID * sizeof(float),
                   hipMemcpyDeviceToDevice, stream);

    for (int t = 0; t < SEQ; ++t) {
        gates_kernel<<<128, 128, 0, stream>>>(x, Wi, Wh, bias, hbuf, act, t);
        cell_kernel<<<(BS * HID) / 256, 256, 0, stream>>>(act, cbuf, hbuf, out, t);
    }

    // final ht, ct appended after output in d_out
    const size_t outElems = (size_t)BS * SEQ * HID;
    hipMemcpyAsync(out + outElems, hbuf, (size_t)BS * HID * sizeof(float),
                   hipMemcpyDeviceToDevice, stream);
    hipMemcpyAsync(out + outElems + (size_t)BS * HID, cbuf,
                   (size_t)BS * HID * sizeof(float),
                   hipMemcpyDeviceToDevice, stream);
}